// Rule_47450798686487
// MI455X (gfx1250) — compile-verified
//
#include <hip/hip_runtime.h>
#include <hip/hip_bf16.h>
#include <cstdint>

#define Bc 4
#define Cc 8
#define Hc 1024
#define Wc 1024
#define TH 16
#define TW 64

__global__ __launch_bounds__(256) void ising_flip_kernel(
    const float* __restrict__ x,     // (B, C+1, H, W)
    const float* __restrict__ rnd,   // (B, C, H, W)
    const float* __restrict__ drp,   // (H, W)
    float* __restrict__ out)         // (B, C+1, H, W)
{
    __shared__ float sT[Cc][TH][TW];        // staged s tile, 32 KB
    __shared__ float Tl[TH + 2][TW + 2];    // channel-sum field + halo

    const int t  = threadIdx.x;
    const int w0 = blockIdx.x * TW;
    const int h0 = blockIdx.y * TH;
    const int b  = blockIdx.z;

    const int r  = t >> 4;          // row in tile: 0..15
    const int cw = (t & 15) << 2;   // col in tile: 0,4,...,60 (float4 aligned)

    const size_t planeHW = (size_t)Hc * Wc;
    const float* xb = x + (size_t)b * (Cc + 1) * planeHW;

    // ---- Phase 1: async-stage interior s tile (8 channels) into LDS ----
    // One GLOBAL_LOAD_ASYNC_TO_LDS_B128 per channel: 256 lanes x 16B = 4 KB tile.
    const size_t rowoff = (size_t)(h0 + r) * Wc + (size_t)(w0 + cw);
#pragma unroll
    for (int c = 0; c < Cc; ++c) {
        uint64_t ga = (uint64_t)(uintptr_t)(xb + (size_t)c * planeHW + rowoff);
        uint32_t la = (uint32_t)(uintptr_t)&sT[c][r][cw];   // low 32 bits = LDS byte addr
        asm volatile("global_load_async_to_lds_b128 %0, %1, off"
                     :: "v"(la), "v"(ga) : "memory");
    }

    // ---- Phase 2: halo ring of T from global (overlaps the async DMA) ----
    const int ringN = 2 * (TW + 2) + 2 * TH;   // 164 ring pixels
    if (t < ringN) {
        int rr, ww;
        if (t < (TW + 2))               { rr = -1;                   ww = t - 1; }
        else if (t < 2 * (TW + 2))      { rr = TH;                   ww = t - (TW + 2) - 1; }
        else if (t < 2 * (TW + 2) + TH) { rr = t - 2 * (TW + 2);     ww = -1; }
        else                            { rr = t - 2 * (TW + 2) - TH; ww = TW; }
        const int gh = (h0 + rr + Hc) & (Hc - 1);
        const int gw = (w0 + ww + Wc) & (Wc - 1);
        float s = 0.0f;
#pragma unroll
        for (int c = 0; c < Cc; ++c)
            s += xb[(size_t)c * planeHW + (size_t)gh * Wc + gw];
        Tl[rr + 1][ww + 1] = s;
    }

    // ---- Wait for this wave's async LDS writes (lane->lane staging),
    //      then interior T from own-lane staged data ----
    asm volatile("s_wait_asynccnt 0x0" ::: "memory");
#pragma unroll
    for (int j = 0; j < 4; ++j) {
        float s = 0.0f;
#pragma unroll
        for (int c = 0; c < Cc; ++c) s += sT[c][r][cw + j];
        Tl[r + 1][cw + 1 + j] = s;
    }
    __syncthreads();   // publish Tl (and sT across waves) for the stencil phase

    // ---- Phase 3: flip update, fully vectorized b128 traffic ----
    const int gh = h0 + r;
    const int gw = w0 + cw;
    const size_t pix = (size_t)gh * Wc + gw;

    const float4 bv4 = *(const float4*)(xb + (size_t)Cc * planeHW + pix);
    const float4 dr4 = *(const float4*)(drp + pix);
    const float bva[4] = {bv4.x, bv4.y, bv4.z, bv4.w};
    const float drv[4] = {dr4.x, dr4.y, dr4.z, dr4.w};

    float Js[4];
    bool  dm[4];
#pragma unroll
    for (int j = 0; j < 4; ++j) {
        Js[j] = Tl[r][cw + 1 + j] + Tl[r + 2][cw + 1 + j]
              + Tl[r + 1][cw + j] + Tl[r + 1][cw + 2 + j];
        dm[j] = drv[j] > 0.5f;
    }

    float* ob = out + (size_t)b * (Cc + 1) * planeHW;
#pragma unroll
    for (int c = 0; c < Cc; ++c) {
        const float4 r4 = *(const float4*)(rnd + ((size_t)(b * Cc + c)) * planeHW + pix);
        const float rv[4] = {r4.x, r4.y, r4.z, r4.w};
        float ov[4];
#pragma unroll
        for (int j = 0; j < 4; ++j) {
            const float s  = sT[c][r][cw + j];
            const float de = 2.0f * s * Js[j];
            const float p  = (de <= 0.0f) ? 1.0f : __expf(-de * bva[j]);
            const float flip = ((rv[j] < p) && dm[j]) ? -1.0f : 1.0f;
            ov[j] = s * flip;
        }
        float4 o4 = make_float4(ov[0], ov[1], ov[2], ov[3]);
        *(float4*)(ob + (size_t)c * planeHW + pix) = o4;
    }
    // channel C = beta field, pass-through
    *(float4*)(ob + (size_t)Cc * planeHW + pix) = bv4;
}

extern "C" void kernel_launch(void* const* d_in, const int* in_sizes, int n_in,
                              void* d_out, int out_size, void* d_ws, size_t ws_size,
                              hipStream_t stream) {
    (void)in_sizes; (void)n_in; (void)out_size; (void)d_ws; (void)ws_size;
    const float* x   = (const float*)d_in[0];
    const float* rnd = (const float*)d_in[1];
    const float* drp = (const float*)d_in[2];
    // d_in[3] = nn_kernel: fixed cross kernel (N+S+E+W, no center), folded into the stencil.
    float* out = (float*)d_out;

    dim3 grid(Wc / TW, Hc / TH, Bc);   // (16, 64, 4)
    ising_flip_kernel<<<grid, 256, 0, stream>>>(x, rnd, drp, out);
}